// PatchCoherentLoss_4406636446082
// MI455X (gfx1250) — compile-verified
//
#include <hip/hip_runtime.h>

typedef float v2f __attribute__((ext_vector_type(2)));
typedef float v8f __attribute__((ext_vector_type(8)));

namespace {
constexpr int   kBatch  = 4;
constexpr int   kCh     = 3;
constexpr int   kImg    = 96;
constexpr int   kOut    = 90;                 // 96 - 7 + 1
constexpr int   kN      = kOut * kOut;        // 8100 patches per image
constexpr int   kTiles  = (kN + 15) / 16;     // 507 tiles of 16
constexpr int   kNpad   = kTiles * 16;        // 8112
constexpr int   kD      = kCh * 7 * 7;        // 147
constexpr int   kDpad   = 148;                // pad K to multiple of 4
constexpr int   kPairs  = kDpad / 2;          // 74 K-pairs (float2 units)
constexpr int   kKSteps = kDpad / 4;          // 37 WMMA K-steps
constexpr float kAlpha  = 0.05f;
constexpr float kInvD   = 1.0f / 147.0f;
constexpr float kBig    = 3.0e38f;
}

// ---------------------------------------------------------------------------
// Patch extraction into K-pair-interleaved column-major layout:
//   float index = ((b*kPairs + p)*kNpad + i)*2 + s , element d = 2p+s.
// Pad (d >= 147 or i >= 8100) is zero-filled so GEMM / sums are unaffected.
// ---------------------------------------------------------------------------
__global__ void extract_patches_kernel(const float* __restrict__ img,
                                       float* __restrict__ pcm, int total) {
  int idx = blockIdx.x * blockDim.x + threadIdx.x;
  if (idx >= total) return;
  int b    = idx / (kDpad * kNpad);
  int rem  = idx - b * (kDpad * kNpad);
  int p    = rem / (2 * kNpad);
  int rem2 = rem - p * (2 * kNpad);
  int i    = rem2 >> 1;
  int s    = rem2 & 1;
  int d    = 2 * p + s;
  float v = 0.0f;
  if (d < kD && i < kN) {
    int c  = d / 49;
    int r  = d - c * 49;
    int ky = r / 7;
    int kx = r - ky * 7;
    int py = i / kOut;
    int px = i - py * kOut;
    v = img[(((size_t)b * kCh + c) * kImg + (py + ky)) * kImg + (px + kx)];
  }
  pcm[idx] = v;
}

// Per-patch sum of squares (pad pairs contribute zero).
__global__ void sumsq_kernel(const float* __restrict__ pcm,
                             float* __restrict__ s2, int total) {
  int idx = blockIdx.x * blockDim.x + threadIdx.x;
  if (idx >= total) return;
  int b = idx / kNpad;
  int i = idx - b * kNpad;
  const v2f* v = (const v2f*)pcm + (size_t)b * kPairs * kNpad + i;
  float s = 0.0f;
  for (int p = 0; p < kPairs; ++p) {
    v2f q = v[(size_t)p * kNpad];
    s += q.x * q.x + q.y * q.y;
  }
  s2[idx] = s;
}

// ---------------------------------------------------------------------------
// Pass 1: row_min[j] = min_i dist[j,i].
// One wave owns a 16-row (j) tile of Y patches (A operand cached in 74 VGPRs)
// and streams all 507 column tiles, 37 x V_WMMA_F32_16X16X4_F32 per tile.
// A layout: lane%16 -> M row, lane/16 selects K-pair {0,1} vs {2,3}.
// B layout: lane%16 -> N col, same K-pair split -> one b64 load per WMMA.
// ---------------------------------------------------------------------------
__global__ __launch_bounds__(256) void pass_rowmin_kernel(
    const float* __restrict__ Xp, const float* __restrict__ Yp,
    const float* __restrict__ x2, const float* __restrict__ y2,
    float* __restrict__ rowMin) {
  int b     = blockIdx.y;
  int wave  = threadIdx.x >> 5;
  int lane  = threadIdx.x & 31;
  int jTile = blockIdx.x * 8 + wave;
  if (jTile >= kTiles) return;               // wave-uniform exit
  int j0  = jTile * 16;
  int l15 = lane & 15;
  int hi  = lane >> 4;                       // 0: K{0,1}, 1: K{2,3}

  const v2f* Xv = (const v2f*)Xp;
  const v2f* Yv = (const v2f*)Yp;
  const float* x2b = x2 + (size_t)b * kNpad;
  const float* y2b = y2 + (size_t)b * kNpad;

  // Preload A: this wave's 16 rows of Y patches, all K.
  v2f a[kKSteps];
  {
    const v2f* aBase = Yv + ((size_t)b * kPairs + hi) * kNpad + (j0 + l15);
#pragma unroll
    for (int t = 0; t < kKSteps; ++t) a[t] = aBase[(size_t)(2 * t) * kNpad];
  }
  float y2r[8];
#pragma unroll
  for (int v = 0; v < 8; ++v) y2r[v] = y2b[j0 + v + 8 * hi];

  float rmin[8];
#pragma unroll
  for (int v = 0; v < 8; ++v) rmin[v] = kBig;

  for (int iTile = 0; iTile < kTiles; ++iTile) {
    int col = iTile * 16 + l15;
    const v2f* bBase = Xv + ((size_t)b * kPairs + hi) * kNpad + col;
    v8f c = {};
#pragma unroll
    for (int t = 0; t < kKSteps; ++t) {
      v2f bb = bBase[(size_t)(2 * t) * kNpad];
      c = __builtin_amdgcn_wmma_f32_16x16x4_f32(
          false, a[t], false, bb, (short)0, c, false, false);
    }
    float x2c = x2b[col];
    float colMask = (col < kN) ? 0.0f : kBig;  // exclude zero-pad columns
#pragma unroll
    for (int v = 0; v < 8; ++v) {
      float dist = (y2r[v] + x2c - 2.0f * c[v]) * kInvD;
      dist = fmaxf(dist, 0.0f) + colMask;
      rmin[v] = fminf(rmin[v], dist);
    }
  }
  // min across the 16 lanes of each half (each half owns rows v / v+8)
#pragma unroll
  for (int v = 0; v < 8; ++v) {
    float m = rmin[v];
    m = fminf(m, __shfl_xor(m, 1, 16));
    m = fminf(m, __shfl_xor(m, 2, 16));
    m = fminf(m, __shfl_xor(m, 4, 16));
    m = fminf(m, __shfl_xor(m, 8, 16));
    int row = j0 + v + 8 * hi;
    if (l15 == 0 && row < kN) rowMin[(size_t)b * kNpad + row] = m;
  }
}

// ---------------------------------------------------------------------------
// Pass 2: col_min[i] = min_j dist[j,i] / (row_min[j] + alpha).
// One wave owns a 16-column (i) tile (B cached in 74 VGPRs), streams j tiles.
// ---------------------------------------------------------------------------
__global__ __launch_bounds__(256) void pass_colmin_kernel(
    const float* __restrict__ Xp, const float* __restrict__ Yp,
    const float* __restrict__ x2, const float* __restrict__ y2,
    const float* __restrict__ rowMin, float* __restrict__ colMin) {
  int b     = blockIdx.y;
  int wave  = threadIdx.x >> 5;
  int lane  = threadIdx.x & 31;
  int iTile = blockIdx.x * 8 + wave;
  if (iTile >= kTiles) return;
  int i0  = iTile * 16;
  int l15 = lane & 15;
  int hi  = lane >> 4;

  const v2f* Xv = (const v2f*)Xp;
  const v2f* Yv = (const v2f*)Yp;
  const float* x2b = x2 + (size_t)b * kNpad;
  const float* y2b = y2 + (size_t)b * kNpad;
  const float* rmb = rowMin + (size_t)b * kNpad;

  // Preload B: this wave's 16 columns of X patches, all K.
  v2f breg[kKSteps];
  {
    const v2f* bBase = Xv + ((size_t)b * kPairs + hi) * kNpad + (i0 + l15);
#pragma unroll
    for (int t = 0; t < kKSteps; ++t) breg[t] = bBase[(size_t)(2 * t) * kNpad];
  }
  float x2c  = x2b[i0 + l15];
  float cmin = kBig;

  for (int jTile = 0; jTile < kTiles; ++jTile) {
    int j0 = jTile * 16;
    const v2f* aBase = Yv + ((size_t)b * kPairs + hi) * kNpad + (j0 + l15);
    v8f c = {};
#pragma unroll
    for (int t = 0; t < kKSteps; ++t) {
      v2f aa = aBase[(size_t)(2 * t) * kNpad];
      c = __builtin_amdgcn_wmma_f32_16x16x4_f32(
          false, aa, false, breg[t], (short)0, c, false, false);
    }
#pragma unroll
    for (int v = 0; v < 8; ++v) {
      int row   = j0 + v + 8 * hi;
      float yv  = y2b[row];
      float rm  = rmb[row];                  // pad rows masked below
      float d   = fmaxf((yv + x2c - 2.0f * c[v]) * kInvD, 0.0f);
      float dn  = d / (rm + kAlpha);
      dn = (row < kN) ? dn : kBig;
      cmin = fminf(cmin, dn);
    }
  }
  // combine row-halves: lane l (rows 0-7 group) with lane l+16 (rows 8-15)
  cmin = fminf(cmin, __shfl_xor(cmin, 16, 32));
  int col = i0 + l15;
  if (hi == 0 && col < kN) colMin[(size_t)b * kNpad + col] = cmin;
}

// Deterministic final reduction: mean over columns per image, mean over batch.
__global__ void finalize_kernel(const float* __restrict__ colMin,
                                float* __restrict__ out) {
  __shared__ float sdata[256];
  int tid = threadIdx.x;
  float total = 0.0f;
  for (int b = 0; b < kBatch; ++b) {
    float s = 0.0f;
    for (int i = tid; i < kN; i += 256) s += colMin[(size_t)b * kNpad + i];
    sdata[tid] = s;
    __syncthreads();
    for (int off = 128; off > 0; off >>= 1) {
      if (tid < off) sdata[tid] += sdata[tid + off];
      __syncthreads();
    }
    if (tid == 0) total += sdata[0] / (float)kN;
    __syncthreads();
  }
  if (tid == 0) out[0] = total / (float)kBatch;
}

extern "C" void kernel_launch(void* const* d_in, const int* in_sizes, int n_in,
                              void* d_out, int out_size, void* d_ws, size_t ws_size,
                              hipStream_t stream) {
  const float* x = (const float*)d_in[0];   // [4,3,96,96] f32
  const float* y = (const float*)d_in[1];   // [4,3,96,96] f32
  float* ws = (float*)d_ws;

  const size_t szPatch = (size_t)kBatch * kDpad * kNpad;  // 4,802,304 floats
  const size_t szVec   = (size_t)kBatch * kNpad;          //    32,448 floats
  float* Xp     = ws;
  float* Yp     = Xp + szPatch;
  float* x2     = Yp + szPatch;
  float* y2     = x2 + szVec;
  float* rowMin = y2 + szVec;
  float* colMin = rowMin + szVec;
  // total ws use: 2*szPatch + 4*szVec floats ~= 37.1 MiB

  const int totalP = (int)szPatch;
  extract_patches_kernel<<<(totalP + 255) / 256, 256, 0, stream>>>(x, Xp, totalP);
  extract_patches_kernel<<<(totalP + 255) / 256, 256, 0, stream>>>(y, Yp, totalP);

  const int totalV = (int)szVec;
  sumsq_kernel<<<(totalV + 255) / 256, 256, 0, stream>>>(Xp, x2, totalV);
  sumsq_kernel<<<(totalV + 255) / 256, 256, 0, stream>>>(Yp, y2, totalV);

  dim3 grid((kTiles + 7) / 8, kBatch);   // 8 waves per block, 1 tile per wave
  pass_rowmin_kernel<<<grid, 256, 0, stream>>>(Xp, Yp, x2, y2, rowMin);
  pass_colmin_kernel<<<grid, 256, 0, stream>>>(Xp, Yp, x2, y2, rowMin, colMin);

  finalize_kernel<<<1, 256, 0, stream>>>(colMin, (float*)d_out);
}